// MovingNCA_39453569581129
// MI455X (gfx1250) — compile-verified
//
#include <hip/hip_runtime.h>
#include <hip/hip_bf16.h>

typedef __attribute__((ext_vector_type(2))) float v2f;
typedef __attribute__((ext_vector_type(8))) float v8f;

#define B_      16
#define NN      64
#define C_      42          // state channels
#define NEO     62
#define CPB     (NEO*NEO)   // 3844 cells per batch image
#define CELLS   (B_*CPB)    // 61504
#define TILES   (CELLS/16)  // 3844 (exact)
#define KREAL   389
#define KPAD    392
#define KK      (KPAD/4)    // 98
#define OUTD    44
#define NPAD    48
#define HID     128
#define TH_     0.0007f
#define SZ_STATE (B_*NN*NN*C_)   // 2752512 floats
#define WC_FLOATS (KPAD*NPAD)    // 18816 floats
#define GUESS_OFF (CELLS*10)     // 615040 floats into d_out

// ---------------- init: copy state0 into both ping-pong buffers, copy pos ---
__global__ void nca_init(const float* __restrict__ state0,
                         const int* __restrict__ perc0,
                         float* __restrict__ stateA, float* __restrict__ stateB,
                         int* __restrict__ pos) {
  int idx = blockIdx.x * blockDim.x + threadIdx.x;
  if (idx < SZ_STATE) { float v = state0[idx]; stateA[idx] = v; stateB[idx] = v; }
  if (idx < CELLS * 2) pos[idx] = perc0[idx];
}

// ---------------- combine the two linear layers: Wc = W1@W2 (swizzled), bc ---
// Wc stored pre-swizzled for the WMMA B operand:
//   float index for (k,n): (((k>>1)*3 + (n>>4))*16 + (n&15))*2 + (k&1)
__global__ void nca_combine(const float* __restrict__ W1, const float* __restrict__ b1,
                            const float* __restrict__ W2, const float* __restrict__ b2,
                            float* __restrict__ Wc, float* __restrict__ bc) {
  int idx = blockIdx.x * blockDim.x + threadIdx.x;
  if (idx < KPAD * NPAD) {
    int k = idx / NPAD, n = idx % NPAD;
    float acc = 0.f;
    if (k < KREAL && n < OUTD) {
      for (int h = 0; h < HID; ++h) acc += W1[k*HID + h] * W2[h*OUTD + n];
    }
    int nt = n >> 4, nl = n & 15;
    Wc[(((k >> 1)*3 + nt)*16 + nl)*2 + (k & 1)] = acc;
  }
  if (idx < NPAD) {
    float acc = 0.f;
    if (idx < OUTD) {
      acc = b2[idx];
      for (int h = 0; h < HID; ++h) acc += b1[h] * W2[h*OUTD + idx];
    }
    bc[idx] = acc;
  }
}

// ---------------- one NCA step: gather -> WMMA GEMM -> state/pos/out update --
__global__ void __launch_bounds__(256)
nca_step(const float* __restrict__ images, const float* __restrict__ stateC,
         float* __restrict__ stateN, int* __restrict__ pos,
         const float* __restrict__ Wc, const float* __restrict__ bc,
         float* __restrict__ dout, int last) {
  extern __shared__ float smem[];
  float* WcS  = smem;                               // 18816 floats (shared by WG)
  float* Xs   = WcS + WC_FLOATS;                    // 8 waves * 16*392 floats
  float* outS = Xs + 8 * 16 * KPAD;                 // 8 waves * 16*48 floats
  int*   meta = (int*)(outS + 8 * 16 * NPAD);       // 8 waves * 16*5 ints

  const int tid  = threadIdx.x;
  const int wave = tid >> 5;
  const int lane = tid & 31;
  const int tile = blockIdx.x * 8 + wave;
  const bool active = (tile < TILES);

  float* myX = Xs   + wave * 16 * KPAD;
  float* myO = outS + wave * 16 * NPAD;
  int*   myM = meta + wave * 16 * 5;

  // per-tile cell metadata (b, i, j, xp, yp) for the 16 cells
  if (active && lane < 16) {
    int cid = tile * 16 + lane;
    int b = cid / CPB, r = cid % CPB;
    int i = r / NEO, j = r % NEO;
    myM[lane*5 + 0] = b;
    myM[lane*5 + 1] = i;
    myM[lane*5 + 2] = j;
    myM[lane*5 + 3] = pos[cid*2 + 0];
    myM[lane*5 + 4] = pos[cid*2 + 1];
  }
  // cooperative load of composed weights (already B-swizzled) into LDS
  for (int idx = tid; idx < WC_FLOATS; idx += 256) WcS[idx] = Wc[idx];
  __syncthreads();

  // ---- gather X tile (16 cells x 392 padded features), WMMA-A swizzled:
  //      float index for (m, f): (f>>1)*32 + m*2 + (f&1)
  if (active) {
    for (int t = 0; t < 13; ++t) {
      int f = t * 32 + lane;
      if (f >= KPAD) break;
      int kind, c = 0, dx = 0, dy = 0;
      if (f < 387) {
        int p = f / 43;             // patch 0..8
        c  = f - p * 43;            // 0 = image, 1..42 = state channel c-1
        dx = p / 3; dy = p - dx * 3;
        kind = (c == 0) ? 0 : 1;
      } else if (f == 387) kind = 2;
      else if (f == 388)   kind = 3;
      else                 kind = 4;
      float* dst = myX + (f >> 1) * 32 + (f & 1);
      for (int m = 0; m < 16; ++m) {
        int b  = myM[m*5 + 0], i = myM[m*5 + 1], j = myM[m*5 + 2];
        int xp = myM[m*5 + 3], yp = myM[m*5 + 4];
        float val;
        if (kind == 0)      val = images[(b*NN + xp + dx)*NN + yp + dy];
        else if (kind == 1) val = stateC[((b*NN + i + dx)*NN + j + dy)*C_ + (c - 1)];
        else if (kind == 2) val = (float)(xp - 32) * (1.0f / 32.0f);
        else if (kind == 3) val = (float)(yp - 32) * (1.0f / 32.0f);
        else                val = 0.0f;
        dst[m * 2] = val;
      }
    }
  }
  __syncthreads();

  // ---- GEMM: [16 x 392] @ [392 x 48] with v_wmma_f32_16x16x4_f32
  v8f acc0 = {}, acc1 = {}, acc2 = {};
  if (active) {
    const v2f* Xv = (const v2f*)myX;
    const v2f* Wv = (const v2f*)WcS;
    const int hi = lane >> 4;     // selects K pair within k-step
    const int nl = lane & 15;     // A: row m / B: column n
    for (int kk = 0; kk < KK; ++kk) {
      v2f a  = Xv[(kk*2 + hi)*16 + nl];
      v2f b0 = Wv[((kk*2 + hi)*3 + 0)*16 + nl];
      v2f b1v= Wv[((kk*2 + hi)*3 + 1)*16 + nl];
      v2f b2v= Wv[((kk*2 + hi)*3 + 2)*16 + nl];
      acc0 = __builtin_amdgcn_wmma_f32_16x16x4_f32(false, a, false, b0, (short)0, acc0, false, false);
      acc1 = __builtin_amdgcn_wmma_f32_16x16x4_f32(false, a, false, b1v, (short)0, acc1, false, false);
      acc2 = __builtin_amdgcn_wmma_f32_16x16x4_f32(false, a, false, b2v, (short)0, acc2, false, false);
    }
    // bias add + spill C tile to LDS (C layout: lane nl = column, vgpr v -> row)
    const float bb0 = bc[nl], bb1 = bc[16 + nl], bb2 = bc[32 + nl];
    #pragma unroll
    for (int v = 0; v < 8; ++v) {
      int m = v + hi * 8;
      myO[m*NPAD +  0 + nl] = acc0[v] + bb0;
      myO[m*NPAD + 16 + nl] = acc1[v] + bb1;
      myO[m*NPAD + 32 + nl] = acc2[v] + bb2;
    }
  }
  __syncthreads();

  // ---- epilogue: state accumulate, outputs on last step, position update
  if (active && lane < 16) {
    const int m  = lane;
    const int cid = tile * 16 + m;
    const int b  = myM[m*5 + 0], i = myM[m*5 + 1], j = myM[m*5 + 2];
    int xp = myM[m*5 + 3], yp = myM[m*5 + 4];
    const long sbase = (((long)b*NN + (i + 1))*NN + (j + 1)) * C_;
    for (int c = 0; c < C_; ++c) {
      float outv = myO[m*NPAD + c];
      int f = 173 + c;                         // center patch (dx=1,dy=1), channel c
      float prev = myX[(f >> 1)*32 + m*2 + (f & 1)];   // stateC value, already in LDS
      float nv = prev + outv;
      stateN[sbase + c] = nv;
      if (last && c >= 32)                     // class_state = last 10 channels
        dout[((b*NEO + i)*NEO + j)*10 + (c - 32)] = nv;
    }
    if (last) {
      for (int c = 0; c < OUTD; ++c)
        dout[GUESS_OFF + (long)cid*OUTD + c] = myO[m*NPAD + c];
    }
    float ax = myO[m*NPAD + 42], ay = myO[m*NPAD + 43];
    int ddx = (ax > TH_) ? 1 : ((ax < -TH_) ? -1 : 0);
    int ddy = (ay > TH_) ? 1 : ((ay < -TH_) ? -1 : 0);
    xp = min(max(xp + ddx, 0), NEO - 1);
    yp = min(max(yp + ddy, 0), NEO - 1);
    pos[cid*2 + 0] = xp;
    pos[cid*2 + 1] = yp;
  }
}

// ---------------------------------------------------------------------------
extern "C" void kernel_launch(void* const* d_in, const int* in_sizes, int n_in,
                              void* d_out, int out_size, void* d_ws, size_t ws_size,
                              hipStream_t stream) {
  const float* images = (const float*)d_in[0];
  const int*   perc0  = (const int*)d_in[1];
  const float* state0 = (const float*)d_in[2];
  const float* W1     = (const float*)d_in[3];
  const float* b1     = (const float*)d_in[4];
  const float* W2     = (const float*)d_in[5];
  const float* b2     = (const float*)d_in[6];
  float* out = (float*)d_out;

  float* stateA = (float*)d_ws;
  float* stateB = stateA + SZ_STATE;
  float* Wc     = stateB + SZ_STATE;
  float* bc     = Wc + WC_FLOATS;
  int*   pos    = (int*)(bc + NPAD);

  nca_init<<<(SZ_STATE + 255)/256, 256, 0, stream>>>(state0, perc0, stateA, stateB, pos);
  nca_combine<<<(KPAD*NPAD + 255)/256, 256, 0, stream>>>(W1, b1, W2, b2, Wc, bc);

  size_t smem = (size_t)(WC_FLOATS + 8*16*KPAD + 8*16*NPAD) * sizeof(float)
              + (size_t)(8*16*5) * sizeof(int);
  int blocks = (TILES + 7) / 8;   // 481
  nca_step<<<blocks, 256, smem, stream>>>(images, stateA, stateB, pos, Wc, bc, out, 0);
  nca_step<<<blocks, 256, smem, stream>>>(images, stateB, stateA, pos, Wc, bc, out, 0);
  nca_step<<<blocks, 256, smem, stream>>>(images, stateA, stateB, pos, Wc, bc, out, 0);
  nca_step<<<blocks, 256, smem, stream>>>(images, stateB, stateA, pos, Wc, bc, out, 1);
}